// Net_22333829939699
// MI455X (gfx1250) — compile-verified
//
#include <hip/hip_runtime.h>
#include <hip/hip_bf16.h>
#include <math.h>

// ---------------------------------------------------------------------------
// Problem constants (from the reference)
// ---------------------------------------------------------------------------
#define BATCH   4096
#define IMG     192
#define HC      64          // conv output height/width (192/3)
#define S_DIM   9
#define HID     15
#define PADW    6

#define STR     20          // LDS row stride in dwords (conflict-free for 16 lanes)

typedef __attribute__((ext_vector_type(2))) float v2f;
typedef __attribute__((ext_vector_type(8))) float v8f;

// ---------------------------------------------------------------------------
// Kernel P: pack the six 9/15-row weight matrices into dense zero-padded
// 16x16 blocks (and biases into padded 16-rows) so the WMMA kernel can load
// fragments with unconditional aligned vector loads (no exec-mask branches).
// Layer order: 0:xw1 1:xw2 2:xw3 3:yw1 4:yw2 5:yw3  (all have in_dim 9).
// ---------------------------------------------------------------------------
__global__ void pack_params_kernel(const float* __restrict__ xw1,
                                   const float* __restrict__ xw2,
                                   const float* __restrict__ xw3,
                                   const float* __restrict__ yw1,
                                   const float* __restrict__ yw2,
                                   const float* __restrict__ yw3,
                                   const float* __restrict__ xb1,
                                   const float* __restrict__ xb2,
                                   const float* __restrict__ xb3,
                                   const float* __restrict__ yb1,
                                   const float* __restrict__ yb2,
                                   const float* __restrict__ yb3,
                                   float* __restrict__ WP,
                                   float* __restrict__ BP) {
    const float* Ws[6] = {xw1, xw2, xw3, yw1, yw2, yw3};
    const float* Bs[6] = {xb1, xb2, xb3, yb1, yb2, yb3};
    const int    od[6] = {S_DIM, S_DIM, HID, S_DIM, S_DIM, HID};
    int tid = threadIdx.x;
    for (int idx = tid; idx < 6 * 256; idx += 256) {
        int l = idx >> 8, e = idx & 255, out = e >> 4, k = e & 15;
        float v = 0.0f;
        if (out < od[l] && k < S_DIM) v = Ws[l][out * S_DIM + k];
        WP[idx] = v;
    }
    for (int idx = tid; idx < 6 * 16; idx += 256) {
        int l = idx >> 4, o = idx & 15;
        BP[idx] = (o < od[l]) ? Bs[l][o] : 0.0f;
    }
}

// ---------------------------------------------------------------------------
// Kernel A: fused conv3x3-stride3 + windowed gather.
// Only the 18 consumed conv outputs per image are computed (vs 4096 in ref).
// ---------------------------------------------------------------------------
__global__ void conv_gather_kernel(const float* __restrict__ x,
                                   const int*   __restrict__ ij,
                                   const float* __restrict__ cw,
                                   const float* __restrict__ cb,
                                   float* __restrict__ P1,
                                   float* __restrict__ P2) {
    int t = blockIdx.x * blockDim.x + threadIdx.x;
    if (t >= BATCH * 18) return;
    int b  = t / 18;
    int p  = t % 18;
    int rr = p / 6;
    int cc = p % 6;
    int i = ij[2 * b + 0];
    int j = ij[2 * b + 1];
    int row = (i - PADW + rr + HC) % HC;
    int col = (j - PADW + cc + HC) % HC;
    const float* xb = x + (size_t)b * IMG * IMG;
    float acc = cb[0];
#pragma unroll
    for (int u = 0; u < 3; ++u)
#pragma unroll
        for (int v = 0; v < 3; ++v)
            acc = fmaf(cw[u * 3 + v], xb[(3 * row + u) * IMG + (3 * col + v)], acc);
    if (cc < 3) P1[b * S_DIM + rr * 3 + cc]     = acc;
    else        P2[b * S_DIM + rr * 3 + cc - 3] = acc;
}

// ---------------------------------------------------------------------------
// WMMA fragment helpers (V_WMMA_F32_16X16X4_F32, wave32).
//   D(out x sample) = W(out x K) * act(K x sample) + C(bias)
// A layout (16x4, ISA 7.12.2): lane = M (out), K = 4c + 2*(lane/16) + r
// B layout (4x16):             lane%16 = N (sample), K = 4c + 2*(lane/16) + r
// C/D layout (16x16):          lane%16 = N, M = r + 8*(lane/16)
// Weights come from the packed zero-padded 16x16 blocks -> plain b64 loads.
// ---------------------------------------------------------------------------
__device__ inline v2f wfrag16(const float* __restrict__ Wp, int c, int lane) {
    int out = lane & 15, g = lane >> 4;
    int k0 = 4 * c + 2 * g;                       // even -> 8B aligned
    return *(const v2f*)(Wp + out * 16 + k0);
}

__device__ inline v2f bfrag(const float* __restrict__ lds, int c, int lane) {
    int s = lane & 15, g = lane >> 4;
    int k0 = 4 * c + 2 * g;
    v2f b;
    b.x = lds[s * STR + k0];
    b.y = lds[s * STR + k0 + 1];
    return b;
}

__device__ inline v8f biasfrag16(const float* __restrict__ Bp, int lane) {
    int g = lane >> 4;
    return *(const v8f*)(Bp + 8 * g);             // 32B aligned vector load
}

// One residual layer: out = in + relu(W*in + b), 9->9 (K padded to 12).
__device__ inline void res_layer(const float* __restrict__ in,
                                 float* __restrict__ out,
                                 const float* __restrict__ Wp,
                                 const float* __restrict__ Bp,
                                 int lane) {
    v8f acc = biasfrag16(Bp, lane);
#pragma unroll
    for (int c = 0; c < 3; ++c)
        acc = __builtin_amdgcn_wmma_f32_16x16x4_f32(
            false, wfrag16(Wp, c, lane),
            false, bfrag(in, c, lane),
            (short)0, acc, false, false);
    int s = lane & 15, g = lane >> 4;
#pragma unroll
    for (int r = 0; r < 8; ++r) {
        int o = r + 8 * g;
        float t = acc[r];
        t = t > 0.0f ? t : 0.0f;
        out[s * STR + o] = in[s * STR + o] + t;
    }
}

// One full branch (9->9->9->15) for a 16-sample tile owned by one wave.
// WPb/BPb point at this branch's 3 packed layers (each 256 / 16 floats).
__device__ inline void mlp_branch(float* __restrict__ buf0,
                                  float* __restrict__ buf1,
                                  const float* __restrict__ P,
                                  const float* __restrict__ WPb,
                                  const float* __restrict__ BPb,
                                  float* __restrict__ OUT,
                                  int sbase, int lane) {
    // Load 16x9 tile into LDS [sample][feature], zero pad features 9..15.
    for (int idx = lane; idx < 16 * 16; idx += 32) {
        int s = idx >> 4, k = idx & 15;
        buf0[s * STR + k] = (k < S_DIM) ? P[(sbase + s) * S_DIM + k] : 0.0f;
    }
    res_layer(buf0, buf1, WPb + 0 * 256, BPb + 0 * 16, lane);
    res_layer(buf1, buf0, WPb + 1 * 256, BPb + 1 * 16, lane);
    // Layer 3: 9 -> 15, no relu / residual; results straight to global.
    v8f acc = biasfrag16(BPb + 2 * 16, lane);
#pragma unroll
    for (int c = 0; c < 3; ++c)
        acc = __builtin_amdgcn_wmma_f32_16x16x4_f32(
            false, wfrag16(WPb + 2 * 256, c, lane),
            false, bfrag(buf0, c, lane),
            (short)0, acc, false, false);
    int s = lane & 15, g = lane >> 4;
#pragma unroll
    for (int r = 0; r < 8; ++r)
        OUT[(size_t)(sbase + s) * 16 + r + 8 * g] = acc[r];
}

// ---------------------------------------------------------------------------
// Kernel B: both branch MLPs via fp32 WMMA. 8 waves/block, 16 samples/wave.
// ---------------------------------------------------------------------------
__global__ void mi_branches_kernel(const float* __restrict__ P1,
                                   const float* __restrict__ P2,
                                   const float* __restrict__ WP,
                                   const float* __restrict__ BP,
                                   float* __restrict__ IX,
                                   float* __restrict__ IY) {
    __shared__ float lds[8 * 2 * 16 * STR];      // 8 waves * ping-pong buffers
    int lane = threadIdx.x & 31;
    int wave = threadIdx.x >> 5;
    int tile = blockIdx.x * 8 + wave;
    int sbase = tile * 16;
    float* buf0 = &lds[wave * 2 * 16 * STR];
    float* buf1 = buf0 + 16 * STR;

    mlp_branch(buf0, buf1, P1, WP,           BP,          IX, sbase, lane);
    mlp_branch(buf0, buf1, P2, WP + 3 * 256, BP + 3 * 16, IY, sbase, lane);
}

// ---------------------------------------------------------------------------
// Kernel C: head (15-dot) for pred_xy and permuted pred_x_y + block reduction.
// ---------------------------------------------------------------------------
__global__ void head_kernel(const float* __restrict__ IX,
                            const float* __restrict__ IY,
                            const int*   __restrict__ perm,
                            const float* __restrict__ fw,
                            const float* __restrict__ fb,
                            float* __restrict__ partials) {
    __shared__ float r1[256];
    __shared__ float r2[256];
    int tid = threadIdx.x;
    int b = blockIdx.x * blockDim.x + tid;
    int pb = perm[b];
    float s1 = fb[0];
    float s2 = fb[0];
#pragma unroll
    for (int k = 0; k < HID; ++k) {
        float ix  = IX[(size_t)b  * 16 + k];
        float iy  = IY[(size_t)b  * 16 + k];
        float iyp = IY[(size_t)pb * 16 + k];
        float h1 = ix + iy;  h1 = h1 > 0.0f ? h1 : 0.0f;
        float h2 = ix + iyp; h2 = h2 > 0.0f ? h2 : 0.0f;
        s1 = fmaf(h1, fw[k], s1);
        s2 = fmaf(h2, fw[k], s2);
    }
    r1[tid] = s1;
    r2[tid] = expf(s2);
    __syncthreads();
    for (int off = 128; off > 0; off >>= 1) {
        if (tid < off) {
            r1[tid] += r1[tid + off];
            r2[tid] += r2[tid + off];
        }
        __syncthreads();
    }
    if (tid == 0) {
        partials[2 * blockIdx.x + 0] = r1[0];
        partials[2 * blockIdx.x + 1] = r2[0];
    }
}

// ---------------------------------------------------------------------------
// Kernel D: final scalar  out = log(mean(exp(pred_x_y))) - mean(pred_xy)
// ---------------------------------------------------------------------------
__global__ void final_kernel(const float* __restrict__ partials, int nblk,
                             float* __restrict__ out) {
    if (threadIdx.x == 0 && blockIdx.x == 0) {
        float s1 = 0.0f, s2 = 0.0f;
        for (int i = 0; i < nblk; ++i) {
            s1 += partials[2 * i + 0];
            s2 += partials[2 * i + 1];
        }
        out[0] = logf(s2 / (float)BATCH) - s1 / (float)BATCH;
    }
}

// ---------------------------------------------------------------------------
// Host launcher
// ---------------------------------------------------------------------------
extern "C" void kernel_launch(void* const* d_in, const int* in_sizes, int n_in,
                              void* d_out, int out_size, void* d_ws, size_t ws_size,
                              hipStream_t stream) {
    const float* x      = (const float*)d_in[0];
    const int*   ij     = (const int*)  d_in[1];
    const int*   perm   = (const int*)  d_in[2];
    const float* conv_w = (const float*)d_in[3];
    const float* conv_b = (const float*)d_in[4];
    const float* xw1    = (const float*)d_in[5];
    const float* xb1    = (const float*)d_in[6];
    const float* xw2    = (const float*)d_in[7];
    const float* xb2    = (const float*)d_in[8];
    const float* xw3    = (const float*)d_in[9];
    const float* xb3    = (const float*)d_in[10];
    const float* yw1    = (const float*)d_in[11];
    const float* yb1    = (const float*)d_in[12];
    const float* yw2    = (const float*)d_in[13];
    const float* yb2    = (const float*)d_in[14];
    const float* yw3    = (const float*)d_in[15];
    const float* yb3    = (const float*)d_in[16];
    const float* fw     = (const float*)d_in[17];
    const float* fb     = (const float*)d_in[18];

    // Workspace layout (floats): WP | BP | P1 | P2 | IX | IY | partials (<1MB)
    float* ws = (float*)d_ws;
    float* WP = ws;                        // 6*256
    float* BP = WP + 6 * 256;              // 6*16
    float* P1 = BP + 6 * 16;               // 4096*9
    float* P2 = P1 + BATCH * S_DIM;        // 4096*9
    float* IX = P2 + BATCH * S_DIM;        // 4096*16
    float* IY = IX + BATCH * 16;           // 4096*16
    float* partials = IY + BATCH * 16;     // 2*nblk

    // P: pack zero-padded weights/biases for branch MLPs (1 block)
    pack_params_kernel<<<1, 256, 0, stream>>>(
        xw1, xw2, xw3, yw1, yw2, yw3,
        xb1, xb2, xb3, yb1, yb2, yb3, WP, BP);

    // A: fused conv + gather (73728 threads)
    conv_gather_kernel<<<(BATCH * 18 + 255) / 256, 256, 0, stream>>>(
        x, ij, conv_w, conv_b, P1, P2);

    // B: WMMA branch MLPs. 256 tiles of 16 samples, 8 waves/block -> 32 blocks.
    mi_branches_kernel<<<BATCH / (16 * 8), 256, 0, stream>>>(
        P1, P2, WP, BP, IX, IY);

    // C: head + per-block reductions (16 blocks of 256)
    int nblk = BATCH / 256;
    head_kernel<<<nblk, 256, 0, stream>>>(IX, IY, perm, fw, fb, partials);

    // D: final scalar
    final_kernel<<<1, 32, 0, stream>>>(partials, nblk, (float*)d_out);
}